// ExampleGNN_36000415875682
// MI455X (gfx1250) — compile-verified
//
#include <hip/hip_runtime.h>
#include <math.h>

#define N_NODES   50000
#define N_EDGES   640000
#define DIM       128
#define DIM_OUT   32
#define NUM_GRAPHS 64

typedef float v2f __attribute__((ext_vector_type(2)));
typedef float v8f __attribute__((ext_vector_type(8)));

// ---------------------------------------------------------------------------
// Degree / normalization:  deg[i] = 1 (self loop) + #incoming edges,
// then in-place dinv[i] = rsqrt(deg[i]).
// ---------------------------------------------------------------------------
__global__ void gcn_init_deg(float* __restrict__ deg, int n) {
  int i = blockIdx.x * blockDim.x + threadIdx.x;
  if (i < n) deg[i] = 1.0f;
}

__global__ void gcn_count_deg(const int* __restrict__ dst, float* __restrict__ deg, int e) {
  int i = blockIdx.x * blockDim.x + threadIdx.x;
  if (i < e) atomicAdd(&deg[dst[i]], 1.0f);
}

__global__ void gcn_rsqrt_deg(float* __restrict__ deg, int n) {
  int i = blockIdx.x * blockDim.x + threadIdx.x;
  if (i < n) deg[i] = rsqrtf(deg[i]);
}

// ---------------------------------------------------------------------------
// WMMA f32 GEMM:  H[M x 128] = (relu? max(A,0):A)[M x 128] @ W[128 x 128]
// One wave -> one 16x16 tile.  Block = 256 threads = 8 waves = one 16-row
// stripe (tn = wave id, tm = blockIdx).  K marched in steps of 4 with
// V_WMMA_F32_16X16X4_F32 (exact f32, matches reference precision).
//
// Wave32 operand layouts (ISA 7.12.2):
//   A 16x4 : lanes 0-15 -> M=lane, K=k0+{0,1};  lanes 16-31 -> M=lane-16, K=k0+{2,3}
//   B 4x16 : lanes 0-15 -> N=lane, K=k0+{0,1};  lanes 16-31 -> N=lane-16, K=k0+{2,3}
//   C/D    : VGPR r: lanes 0-15 -> (M=r, N=lane); lanes 16-31 -> (M=8+r, N=lane-16)
// ---------------------------------------------------------------------------
__global__ __launch_bounds__(256)
void gcn_gemm_wmma(const float* __restrict__ A, const float* __restrict__ W,
                   float* __restrict__ H, int relu) {
  const int lane  = threadIdx.x & 31;
  const int wid   = threadIdx.x >> 5;
  const int tm    = blockIdx.x;            // 0..3124  (50000/16 tiles of rows)
  const int tn    = wid;                   // 0..7     (128/16 tiles of cols)
  const int l15   = lane & 15;
  const int mrow  = tm * 16 + l15;
  const int ncol  = tn * 16 + l15;
  const int khalf = (lane >> 4) << 1;      // 0 for lanes 0-15, 2 for lanes 16-31

  const float* __restrict__ arow = A + (size_t)mrow * DIM;

  v8f acc = {0.f, 0.f, 0.f, 0.f, 0.f, 0.f, 0.f, 0.f};

#pragma unroll
  for (int k0 = 0; k0 < DIM; k0 += 4) {
    float a0 = arow[k0 + khalf];
    float a1 = arow[k0 + khalf + 1];
    if (relu) { a0 = fmaxf(a0, 0.f); a1 = fmaxf(a1, 0.f); }
    v2f a; a[0] = a0; a[1] = a1;
    v2f b;
    b[0] = W[(size_t)(k0 + khalf)     * DIM + ncol];
    b[1] = W[(size_t)(k0 + khalf + 1) * DIM + ncol];
    acc = __builtin_amdgcn_wmma_f32_16x16x4_f32(
        /*neg_a=*/false, a, /*neg_b=*/false, b,
        /*c_mod=*/(short)0, acc, /*reuse_a=*/false, /*reuse_b=*/false);
  }

  const int orow = tm * 16 + ((lane >> 4) << 3);   // +0 or +8
#pragma unroll
  for (int r = 0; r < 8; ++r)
    H[(size_t)(orow + r) * DIM + ncol] = acc[r];
}

// ---------------------------------------------------------------------------
// out[i][f] = bias[f] + h[i][f] * dinv[i]^2   (self-loop term + bias init)
// One thread per (node, 4 features); float4 traffic.
// ---------------------------------------------------------------------------
__global__ void gcn_self_bias(const float* __restrict__ h, const float* __restrict__ dinv,
                              const float* __restrict__ bias, float* __restrict__ out, int n) {
  int t = blockIdx.x * blockDim.x + threadIdx.x;
  if (t >= n * 32) return;
  int i = t >> 5;
  int f = (t & 31) << 2;
  float d = dinv[i];
  float w = d * d;
  float4 hv = *(const float4*)(h + (size_t)i * DIM + f);
  float4 bv = *(const float4*)(bias + f);
  float4 o;
  o.x = bv.x + hv.x * w;
  o.y = bv.y + hv.y * w;
  o.z = bv.z + hv.z * w;
  o.w = bv.w + hv.w * w;
  *(float4*)(out + (size_t)i * DIM + f) = o;
}

// ---------------------------------------------------------------------------
// Edge scatter:  out[dst] += dinv[src]*dinv[dst] * h[src]
// Thread = (edge, 4 features). Consecutive lanes share an edge -> coalesced
// float4 gather + 4 global_atomic_add_f32 (h/out are L2 resident).
// ---------------------------------------------------------------------------
__global__ void gcn_scatter(const int* __restrict__ src, const int* __restrict__ dst,
                            const float* __restrict__ dinv, const float* __restrict__ h,
                            float* __restrict__ out, int e) {
  long long t = (long long)blockIdx.x * blockDim.x + threadIdx.x;
  if (t >= (long long)e * 32) return;
  int ei = (int)(t >> 5);
  int f  = ((int)t & 31) << 2;
  int s = src[ei];
  int d = dst[ei];
  float w = dinv[s] * dinv[d];
  float4 hv = *(const float4*)(h + (size_t)s * DIM + f);
  float* o = out + (size_t)d * DIM + f;
  atomicAdd(o + 0, hv.x * w);
  atomicAdd(o + 1, hv.y * w);
  atomicAdd(o + 2, hv.z * w);
  atomicAdd(o + 3, hv.w * w);
}

__global__ void gcn_zero(float* __restrict__ p, int n) {
  int i = blockIdx.x * blockDim.x + threadIdx.x;
  if (i < n) p[i] = 0.f;
}

// ---------------------------------------------------------------------------
// global_add_pool of relu(x) with sorted batch ids: thread f accumulates a
// running per-graph sum in a register and flushes with one atomic per
// (graph, block) boundary instead of per node.
// ---------------------------------------------------------------------------
__global__ __launch_bounds__(128)
void gcn_pool(const float* __restrict__ x, const int* __restrict__ batch,
              float* __restrict__ pooled, int n) {
  int f = threadIdx.x;                                   // 0..127
  int per = (n + gridDim.x - 1) / gridDim.x;
  int start = blockIdx.x * per;
  int end = min(start + per, n);
  float acc = 0.f;
  int cur = -1;
  for (int i = start; i < end; ++i) {
    int g = batch[i];
    if (g != cur) {
      if (cur >= 0) atomicAdd(&pooled[(size_t)cur * DIM + f], acc);
      acc = 0.f;
      cur = g;
    }
    acc += fmaxf(x[(size_t)i * DIM + f], 0.f);
  }
  if (cur >= 0) atomicAdd(&pooled[(size_t)cur * DIM + f], acc);
}

// ---------------------------------------------------------------------------
// Head: logits = pooled @ Wh + bh; log_softmax via wave32 shuffle reductions.
// One wave per graph, one lane per class (32 classes == 32 lanes).
// ---------------------------------------------------------------------------
__global__ __launch_bounds__(32)
void gcn_head(const float* __restrict__ pooled, const float* __restrict__ Wh,
              const float* __restrict__ bh, float* __restrict__ out) {
  int g = blockIdx.x;         // 0..63
  int c = threadIdx.x;        // 0..31
  const float* p = pooled + (size_t)g * DIM;
  float s = bh[c];
#pragma unroll 4
  for (int k = 0; k < DIM; ++k) s += p[k] * Wh[(size_t)k * DIM_OUT + c];

  float m = s;
  for (int off = 16; off > 0; off >>= 1) m = fmaxf(m, __shfl_xor(m, off, 32));
  float ex = expf(s - m);
  float sum = ex;
  for (int off = 16; off > 0; off >>= 1) sum += __shfl_xor(sum, off, 32);
  out[(size_t)g * DIM_OUT + c] = s - m - logf(sum);
}

// ---------------------------------------------------------------------------
extern "C" void kernel_launch(void* const* d_in, const int* in_sizes, int n_in,
                              void* d_out, int out_size, void* d_ws, size_t ws_size,
                              hipStream_t stream) {
  const float* x     = (const float*)d_in[0];
  const int*   edge  = (const int*)d_in[1];     // [2, E] flat: row0=src, row1=dst
  const int*   batch = (const int*)d_in[2];
  const float* W1    = (const float*)d_in[3];
  const float* b1    = (const float*)d_in[4];
  const float* W2    = (const float*)d_in[5];
  const float* b2    = (const float*)d_in[6];
  const float* Wh    = (const float*)d_in[7];
  const float* bh    = (const float*)d_in[8];
  float*       out   = (float*)d_out;

  const int* src = edge;
  const int* dst = edge + N_EDGES;

  // Workspace layout (floats), all 16B aligned: dinv | h | buf | pooled
  float* ws     = (float*)d_ws;
  float* dinv   = ws;
  float* h      = ws + 50048;                         // 50000 rounded up
  float* buf    = h + (size_t)N_NODES * DIM;          // 6.4M
  float* pooled = buf + (size_t)N_NODES * DIM;        // 8192

  const int TB = 256;

  // ---- degree / symmetric normalization -------------------------------
  gcn_init_deg <<<(N_NODES + TB - 1) / TB, TB, 0, stream>>>(dinv, N_NODES);
  gcn_count_deg<<<(N_EDGES + TB - 1) / TB, TB, 0, stream>>>(dst, dinv, N_EDGES);
  gcn_rsqrt_deg<<<(N_NODES + TB - 1) / TB, TB, 0, stream>>>(dinv, N_NODES);

  // ---- layer 1:  h = x @ W1 ;  buf = b1 + scatter(norm * h) ------------
  gcn_gemm_wmma<<<N_NODES / 16, TB, 0, stream>>>(x, W1, h, /*relu=*/0);
  gcn_self_bias<<<(N_NODES * 32 + TB - 1) / TB, TB, 0, stream>>>(h, dinv, b1, buf, N_NODES);
  gcn_scatter  <<<(int)(((long long)N_EDGES * 32 + TB - 1) / TB), TB, 0, stream>>>(
      src, dst, dinv, h, buf, N_EDGES);

  // ---- layer 2:  h = relu(buf) @ W2 ;  buf = b2 + scatter(norm * h) ----
  gcn_gemm_wmma<<<N_NODES / 16, TB, 0, stream>>>(buf, W2, h, /*relu=*/1);
  gcn_self_bias<<<(N_NODES * 32 + TB - 1) / TB, TB, 0, stream>>>(h, dinv, b2, buf, N_NODES);
  gcn_scatter  <<<(int)(((long long)N_EDGES * 32 + TB - 1) / TB), TB, 0, stream>>>(
      src, dst, dinv, h, buf, N_EDGES);

  // ---- pool relu(buf) by sorted batch, then head + log_softmax ---------
  gcn_zero<<<(NUM_GRAPHS * DIM + TB - 1) / TB, TB, 0, stream>>>(pooled, NUM_GRAPHS * DIM);
  gcn_pool<<<256, 128, 0, stream>>>(buf, batch, pooled, N_NODES);
  gcn_head<<<NUM_GRAPHS, 32, 0, stream>>>(pooled, Wh, bh, out);
}